// _PNET_post_82841329205498
// MI455X (gfx1250) — compile-verified
//
#include <hip/hip_runtime.h>
#include <math.h>

// ---------------- problem constants ----------------
#define BATCH   16
#define HH      540
#define WW      960
#define NPIX    (HH * WW)           // 518400 pixels per batch
#define RC      12                  // row-chunks per column
#define ROWS    (HH / RC)           // 45 rows per granule
#define GRB     (WW * RC)           // granules per batch = 11520
#define CNT_BLK 256
#define BLKS_PER_B (GRB / CNT_BLK)  // 45 blocks per batch
#define SCAN_THREADS 512            // exactly 16 wave32 waves
#define SCAN_CH 23                  // ceil(11520/512)
#define THRESH  0.6f

static_assert(HH % RC == 0, "rows divide");
static_assert(GRB % CNT_BLK == 0, "granules divide block");
static_assert(SCAN_THREADS * SCAN_CH >= GRB, "scan coverage");

typedef __attribute__((ext_vector_type(2))) float v2f;
typedef __attribute__((ext_vector_type(8))) float v8f;

// stride = (2*960 + 11 - 12) / (960 - 1), computed in double then cast to f32
// to match Python-float -> jnp.float32 exactly.
__device__ __forceinline__ float pnet_stride() { return (float)(1919.0 / 959.0); }

// ---------------- kernel 1: per-granule mask counts ----------------
// Thread handles granule (b, column i, row-chunk r): 45 rows at fixed i.
// Consecutive lanes -> consecutive i -> coalesced (stride-2 float reads).
__global__ void k_count(const float* __restrict__ cls, unsigned* __restrict__ counts) {
    int b = blockIdx.x / BLKS_PER_B;
    int t = (blockIdx.x % BLKS_PER_B) * CNT_BLK + threadIdx.x;  // 0..GRB-1
    int i = t % WW;
    int r = t / WW;
    const float* p = cls + (((size_t)(b * HH + r * ROWS) * WW + i) * 2 + 1);
    unsigned c = 0;
    for (int j = 0; j < ROWS; ++j) {
        c += (p[(size_t)j * WW * 2] >= THRESH) ? 1u : 0u;
    }
    // scan order index is k-order: granule (i, r) -> i*RC + r
    counts[b * GRB + i * RC + r] = c;
}

// ---------------- kernel 2: per-batch exclusive scan ----------------
// 512 threads = 16 waves. per-thread serial sum -> wave32 shuffle inclusive
// scan -> 16-wave totals scanned with 4 chained V_WMMA_F32_16X16X4_F32:
//   D(m,n) = sum_c sum_k [4c+k <= m] * waveTot[4c+k]  =  inclusive scan s_m.
__global__ void k_scan(const unsigned* __restrict__ counts,
                       unsigned* __restrict__ offsets,
                       unsigned* __restrict__ totals) {
    int b    = blockIdx.x;
    int t    = threadIdx.x;
    int lane = t & 31;
    int wv   = t >> 5;                 // 0..15

    __shared__ float waveTot[16];
    __shared__ float S[16];

    const unsigned* cnt = counts + b * GRB;
    int g0 = t * SCAN_CH;

    unsigned p = 0;
    for (int u = 0; u < SCAN_CH; ++u) {
        int g = g0 + u;
        if (g < GRB) p += cnt[g];
    }

    // wave32 inclusive scan of per-thread partials
    unsigned inc = p;
    #pragma unroll
    for (int d = 1; d < 32; d <<= 1) {
        unsigned v = __shfl_up(inc, d, 32);
        if (lane >= d) inc += v;
    }
    if (lane == 31) waveTot[wv] = (float)inc;   // wave total (<= ~33K, exact f32)
    __syncthreads();

    // --- WMMA 16-scan of the 16 wave totals ---
    // A (16x4 f32) layout: lanes 0-15 hold K=0 (v0), K=1 (v1); lanes 16-31 hold K=2,3.
    // B (4x16 f32) layout mirrors: v0 = row K=0 (lanes 0-15) / K=2 (lanes 16-31), v1 = K=1 / K=3.
    int M   = lane & 15;
    int klo = (lane < 16) ? 0 : 2;
    v8f acc = {};
    #pragma unroll
    for (int c = 0; c < 4; ++c) {
        v2f a, bb;
        a.x  = ((4 * c + klo)     <= M) ? 1.0f : 0.0f;
        a.y  = ((4 * c + klo + 1) <= M) ? 1.0f : 0.0f;
        bb.x = waveTot[4 * c + klo];
        bb.y = waveTot[4 * c + klo + 1];
        acc = __builtin_amdgcn_wmma_f32_16x16x4_f32(
            /*neg_a=*/false, a, /*neg_b=*/false, bb,
            /*c_mod=*/(short)0, acc, /*reuse_a=*/false, /*reuse_b=*/false);
    }
    // C/D layout: lanes 0-15 VGPR r = row M=r; lanes 16-31 VGPR r = row M=8+r.
    if (t == 0) {
        #pragma unroll
        for (int r2 = 0; r2 < 8; ++r2) S[r2] = acc[r2];
    }
    if (t == 16) {
        #pragma unroll
        for (int r2 = 0; r2 < 8; ++r2) S[8 + r2] = acc[r2];
    }
    __syncthreads();

    unsigned waveBase = (wv > 0) ? (unsigned)(S[wv - 1] + 0.5f) : 0u;
    unsigned base = waveBase + (inc - p);      // exclusive base for this thread
    for (int u = 0; u < SCAN_CH; ++u) {
        int g = g0 + u;
        if (g < GRB) {
            offsets[b * GRB + g] = base;
            base += cnt[g];
        }
    }
    if (t == 0) totals[b] = (unsigned)(S[15] + 0.5f);
}

// ---------------- kernel 3: compute rects + stable scatter ----------------
__global__ void k_scatter(const float* __restrict__ cls,
                          const float* __restrict__ bbx,
                          const unsigned* __restrict__ offsets,
                          float* __restrict__ out) {
    int b = blockIdx.x / BLKS_PER_B;
    int t = (blockIdx.x % BLKS_PER_B) * CNT_BLK + threadIdx.x;
    int i = t % WW;
    int r = t / WW;
    int j0 = r * ROWS;

    const float stride = pnet_stride();
    const float* pc = cls + (((size_t)(b * HH + j0) * WW + i) * 2);
    const float* pb = bbx + (((size_t)(b * HH + j0) * WW + i) * 4);

    unsigned m = offsets[b * GRB + i * RC + r];

    // column-constant pieces (tffix == trunc for non-negative args)
    float sx  = stride * (float)i;
    float x1  = truncf(sx);
    float sx2 = sx + 11.0f;
    float x2  = truncf(sx2);

    float* o1 = out + (size_t)b * NPIX * 5;
    float* o2 = o1 + (size_t)BATCH * NPIX * 5;

    for (int j = 0; j < ROWS; ++j) {
        if ((j & 7) == 0 && j + 8 < ROWS) {
            __builtin_prefetch(pc + (size_t)(j + 8) * WW * 2, 0, 1);
            __builtin_prefetch(pb + (size_t)(j + 8) * WW * 4, 0, 1);
        }
        float prob = pc[(size_t)j * WW * 2 + 1];
        if (prob >= THRESH) {
            float jj = (float)(j0 + j);
            float sy  = stride * jj;            // separate stmts: no FMA contraction
            float y1  = truncf(sy);
            float sy2 = sy + 11.0f;
            float y2  = truncf(sy2);
            const float* ob = pb + (size_t)j * WW * 4;
            float t0 = ob[0] * 12.0f;  float b0 = x1 + t0;
            float t1 = ob[1] * 12.0f;  float b1 = y1 + t1;
            float t2 = ob[2] * 12.0f;  float b2 = x2 + t2;
            float t3 = ob[3] * 12.0f;  float b3 = y2 + t3;
            float w = b2 - b0;
            float h = b3 - b1;
            float l = fmaxf(w, h);
            float wh = w * 0.5f;  float lh = l * 0.5f;
            float nx1a = b0 + wh;  float nx1 = nx1a - lh;
            float ny1a = b1 + h * 0.5f;  float ny1 = ny1a - lh;
            float nx2 = nx1 + l;
            float ny2 = ny1 + l;
            size_t row = (size_t)m * 5;
            o1[row + 0] = nx1; o1[row + 1] = ny1; o1[row + 2] = nx2;
            o1[row + 3] = ny2; o1[row + 4] = prob;
            o2[row + 0] = nx1; o2[row + 1] = ny1; o2[row + 2] = nx2;
            o2[row + 3] = ny2; o2[row + 4] = prob;
            ++m;
        }
    }
}

// ---------------- kernel 4: zero only the tail [total_b, NPIX) ----------------
__global__ void k_tail(const unsigned* __restrict__ totals, float* __restrict__ out) {
    size_t idx = (size_t)blockIdx.x * blockDim.x + threadIdx.x;
    if (idx >= (size_t)BATCH * NPIX) return;
    int b       = (int)(idx / NPIX);
    unsigned m  = (unsigned)(idx % NPIX);
    if (m < totals[b]) return;
    float* o1 = out + (size_t)b * NPIX * 5 + (size_t)m * 5;
    float* o2 = o1 + (size_t)BATCH * NPIX * 5;
    #pragma unroll
    for (int c = 0; c < 5; ++c) { o1[c] = 0.0f; o2[c] = 0.0f; }
}

// ---------------- kernel 5: img_shape + 1 ----------------
__global__ void k_img(const float* __restrict__ img, float* __restrict__ out) {
    int t = threadIdx.x;
    if (t < BATCH * 2) out[(size_t)2 * BATCH * NPIX * 5 + t] = img[t] + 1.0f;
}

// ---------------- launcher ----------------
extern "C" void kernel_launch(void* const* d_in, const int* in_sizes, int n_in,
                              void* d_out, int out_size, void* d_ws, size_t ws_size,
                              hipStream_t stream) {
    const float* cls = (const float*)d_in[0];   // (16,540,960,2) f32
    const float* bbx = (const float*)d_in[1];   // (16,540,960,4) f32
    const float* img = (const float*)d_in[2];   // (16,2) f32
    float* out = (float*)d_out;

    unsigned* counts  = (unsigned*)d_ws;                       // 16*11520 u32
    unsigned* offsets = counts + (size_t)BATCH * GRB;          // 16*11520 u32
    unsigned* totals  = offsets + (size_t)BATCH * GRB;         // 16 u32

    k_count  <<<BATCH * BLKS_PER_B, CNT_BLK,      0, stream>>>(cls, counts);
    k_scan   <<<BATCH,              SCAN_THREADS, 0, stream>>>(counts, offsets, totals);
    k_scatter<<<BATCH * BLKS_PER_B, CNT_BLK,      0, stream>>>(cls, bbx, offsets, out);

    size_t nrows = (size_t)BATCH * NPIX;
    k_tail   <<<(unsigned)((nrows + 255) / 256), 256, 0, stream>>>(totals, out);
    k_img    <<<1, 32, 0, stream>>>(img, out);
}